// PolyDeconvLoss_7447473292150
// MI455X (gfx1250) — compile-verified
//
#include <hip/hip_runtime.h>
#include <hip/hip_bf16.h>

typedef __attribute__((ext_vector_type(2))) float v2f;
typedef __attribute__((ext_vector_type(8))) float v8f;

#define ROWS        48          // N*C
#define ROW_ELEMS   262144      // H*W
#define ROW_ELEMS4  65536       // H*W / 4
#define BINS        256
#define BPR         16          // blocks per row per tensor
#define NHIST       (2 * ROWS * BINS)   // 24576
#define WS_WORDS    (192 + NHIST)       // 96 min + 96 max + counts

// ---- monotone float <-> ordered uint (so atomicMin/Max on uints == float min/max) ----
__device__ __forceinline__ unsigned f2ord(float f) {
    unsigned u = __float_as_uint(f);
    return (u & 0x80000000u) ? ~u : (u | 0x80000000u);
}
__device__ __forceinline__ float ord2f(unsigned u) {
    return __uint_as_float((u & 0x80000000u) ? (u & 0x7FFFFFFFu) : ~u);
}

// ---------------- 0) workspace init ----------------
__global__ void hloss_init_kernel(unsigned* __restrict__ ws) {
    int i = blockIdx.x * blockDim.x + threadIdx.x;
    if (i < WS_WORDS) {
        // first 96 words: per-(tensor,row) mins -> init to UINT_MAX; rest -> 0
        ws[i] = (i < 96) ? 0xFFFFFFFFu : 0u;
    }
}

// ---------------- 1) per-row min/max ----------------
__global__ void hloss_minmax_kernel(const float* __restrict__ s,
                                    const float* __restrict__ t,
                                    unsigned* __restrict__ umin,
                                    unsigned* __restrict__ umax) {
    __shared__ float smn[256];
    __shared__ float smx[256];

    const int row = blockIdx.y;
    const int z   = blockIdx.z;
    const float* src = (z == 0) ? s : t;

    const int per_seg = ROW_ELEMS4 / BPR;   // 4096 float4 per block
    const float4* base = (const float4*)src + (size_t)row * ROW_ELEMS4
                         + (size_t)blockIdx.x * per_seg;

    float mn =  3.402823466e+38f;
    float mx = -3.402823466e+38f;
    for (int i = threadIdx.x; i < per_seg; i += blockDim.x) {
        float4 v = base[i];                  // global_load_b128
        mn = fminf(mn, fminf(fminf(v.x, v.y), fminf(v.z, v.w)));
        mx = fmaxf(mx, fmaxf(fmaxf(v.x, v.y), fmaxf(v.z, v.w)));
    }
    smn[threadIdx.x] = mn;
    smx[threadIdx.x] = mx;
    __syncthreads();
    for (int off = 128; off > 0; off >>= 1) {
        if (threadIdx.x < off) {
            smn[threadIdx.x] = fminf(smn[threadIdx.x], smn[threadIdx.x + off]);
            smx[threadIdx.x] = fmaxf(smx[threadIdx.x], smx[threadIdx.x + off]);
        }
        __syncthreads();
    }
    if (threadIdx.x == 0) {
        atomicMin(&umin[z * ROWS + row], f2ord(smn[0]));
        atomicMax(&umax[z * ROWS + row], f2ord(smx[0]));
    }
}

// ---------------- 2) per-row 256-bin histogram (LDS privatized) ----------------
__global__ void hloss_hist_kernel(const float* __restrict__ s,
                                  const float* __restrict__ t,
                                  const unsigned* __restrict__ umin,
                                  const unsigned* __restrict__ umax,
                                  unsigned* __restrict__ counts) {
    __shared__ unsigned h[BINS];

    const int row = blockIdx.y;
    const int z   = blockIdx.z;
    const float* src = (z == 0) ? s : t;

    if (threadIdx.x < BINS) h[threadIdx.x] = 0u;
    __syncthreads();

    const float mn = ord2f(umin[z * ROWS + row]);
    const float mx = ord2f(umax[z * ROWS + row]);
    const float rng = (mx > mn) ? (mx - mn) : 1.0f;
    const float scale = (float)BINS / rng;

    const int per_seg = ROW_ELEMS4 / BPR;
    const float4* base = (const float4*)src + (size_t)row * ROW_ELEMS4
                         + (size_t)blockIdx.x * per_seg;

    for (int i = threadIdx.x; i < per_seg; i += blockDim.x) {
        float4 v = base[i];
        int b0 = (int)floorf((v.x - mn) * scale);
        int b1 = (int)floorf((v.y - mn) * scale);
        int b2 = (int)floorf((v.z - mn) * scale);
        int b3 = (int)floorf((v.w - mn) * scale);
        b0 = b0 < 0 ? 0 : (b0 > BINS - 1 ? BINS - 1 : b0);
        b1 = b1 < 0 ? 0 : (b1 > BINS - 1 ? BINS - 1 : b1);
        b2 = b2 < 0 ? 0 : (b2 > BINS - 1 ? BINS - 1 : b2);
        b3 = b3 < 0 ? 0 : (b3 > BINS - 1 ? BINS - 1 : b3);
        atomicAdd(&h[b0], 1u);   // ds_add_u32
        atomicAdd(&h[b1], 1u);
        atomicAdd(&h[b2], 1u);
        atomicAdd(&h[b3], 1u);
    }
    __syncthreads();
    if (threadIdx.x < BINS) {
        atomicAdd(&counts[(size_t)z * ROWS * BINS + row * BINS + threadIdx.x],
                  h[threadIdx.x]);
    }
}

// ---------------- 3) final MSE reduction via v_wmma_f32_16x16x4_f32 ----------------
// One wave (32 lanes). D += ones(16x4) x B(4x16); with A == 1, every element of
// row 0 of D is a column-sum, so summing D[0][0..15] (VGPR0 on lanes 0..15)
// yields the grand total of all B elements fed in. 12288 values / 64 per WMMA
// = 192 iterations.
__global__ void hloss_wmma_loss_kernel(const unsigned* __restrict__ counts,
                                       float* __restrict__ out) {
    const int lane = threadIdx.x;        // 0..31, EXEC all ones, uniform loop
    v2f a;
    a[0] = 1.0f;
    a[1] = 1.0f;
    v8f c = {};

    const float inv_bins = 1.0f / (float)BINS;
    const int total_vals = ROWS * BINS;          // 12288
    for (int it = 0; it < total_vals / 64; ++it) {
        int i0 = it * 64 + lane * 2;
        float d0 = ((float)counts[i0]     - (float)counts[total_vals + i0])     * inv_bins;
        float d1 = ((float)counts[i0 + 1] - (float)counts[total_vals + i0 + 1]) * inv_bins;
        v2f b;
        b[0] = d0 * d0;
        b[1] = d1 * d1;
        c = __builtin_amdgcn_wmma_f32_16x16x4_f32(
                /*neg_a=*/false, a, /*neg_b=*/false, b,
                /*c_mod=*/(short)0, c, /*reuse_a=*/false, /*reuse_b=*/false);
    }

    // Sum D[0][j], j=0..15: VGPR0 (c[0]) of lanes 0..15. XOR masks 1,2,4,8 keep
    // the reduction inside the 16-lane group.
    float rsum = c[0];
    rsum += __shfl_xor(rsum, 1, 32);
    rsum += __shfl_xor(rsum, 2, 32);
    rsum += __shfl_xor(rsum, 4, 32);
    rsum += __shfl_xor(rsum, 8, 32);
    if (lane == 0) {
        out[0] = rsum * (1.0f / (float)(ROWS * BINS));
    }
}

extern "C" void kernel_launch(void* const* d_in, const int* in_sizes, int n_in,
                              void* d_out, int out_size, void* d_ws, size_t ws_size,
                              hipStream_t stream) {
    const float* s = (const float*)d_in[0];
    const float* t = (const float*)d_in[1];
    float* out = (float*)d_out;

    unsigned* ws     = (unsigned*)d_ws;
    unsigned* umin   = ws;          // 2*48 = 96 words
    unsigned* umax   = ws + 96;     // 96 words
    unsigned* counts = ws + 192;    // 2*48*256 = 24576 words

    hloss_init_kernel<<<(WS_WORDS + 255) / 256, 256, 0, stream>>>(ws);

    dim3 grid(BPR, ROWS, 2);
    hloss_minmax_kernel<<<grid, 256, 0, stream>>>(s, t, umin, umax);
    hloss_hist_kernel<<<grid, 256, 0, stream>>>(s, t, umin, umax, counts);
    hloss_wmma_loss_kernel<<<1, 32, 0, stream>>>(counts, out);
}